// depth_channel_att_glo_60060822667452
// MI455X (gfx1250) — compile-verified
//
#include <hip/hip_runtime.h>
#include <math.h>
#include <stdint.h>

typedef __attribute__((ext_vector_type(8))) float v8f;
typedef __attribute__((ext_vector_type(2))) float v2f;

#define DIM   104
#define CR    8
#define GRP   13
#define HW    65536      // 256*256
#define NB    8
#define XS    132        // padded LDS row stride (floats): +4 bank offset between rows
#define TPIX  128        // pixels per workgroup
#define MT    7          // 7 tiles of 16 rows cover 104 (pad to 112)

#if defined(__gfx1250__) && __has_builtin(__builtin_amdgcn_global_load_async_to_lds_b128) && __has_builtin(__builtin_amdgcn_s_wait_asynccnt)
#define HAVE_ASYNC_LDS 1
typedef int v4i __attribute__((vector_size(16)));
typedef __attribute__((address_space(1))) v4i as1_v4i;
typedef __attribute__((address_space(3))) v4i as3_v4i;
#endif

// ---------------- Kernel 1: per-channel-plane sums (for GAP) ----------------
__global__ __launch_bounds__(256) void chsum_kernel(const float* __restrict__ x,
                                                    float* __restrict__ chsum) {
    const int plane = blockIdx.x;                  // 0 .. 8*104-1
    const float4* p = reinterpret_cast<const float4*>(x + (size_t)plane * HW);
    float s = 0.f;
    for (int i = threadIdx.x; i < HW / 4; i += 256) {
        float4 v = p[i];
        s += v.x + v.y + v.z + v.w;
    }
    __shared__ float red[256];
    red[threadIdx.x] = s;
    __syncthreads();
    for (int st = 128; st > 0; st >>= 1) {
        if (threadIdx.x < st) red[threadIdx.x] += red[threadIdx.x + st];
        __syncthreads();
    }
    if (threadIdx.x == 0) chsum[plane] = red[0];
}

// ------- Kernel 2: GAP -> MLP -> BN -> sigmoid -> fold F_b, gamma into A ----
__global__ __launch_bounds__(256) void filt_kernel(const float* __restrict__ chsum,
                                                   const float* __restrict__ dw,
                                                   const float* __restrict__ cw,
                                                   const float* __restrict__ bnw,
                                                   const float* __restrict__ bnb,
                                                   const float* __restrict__ bnm,
                                                   const float* __restrict__ bnv,
                                                   const float* __restrict__ pw,
                                                   const float* __restrict__ gamma,
                                                   float* __restrict__ Apad) {
    __shared__ float gl[NB * CR];        // g[b][gr]
    __shared__ float fl[NB * 24];        // sigmoid outputs f[b][k*8+c]
    const int t = threadIdx.x;
    if (t < NB * CR) {
        int b = t >> 3, gr = t & 7;
        float s = 0.f;
        #pragma unroll
        for (int i = 0; i < GRP; ++i) s += dw[gr * GRP + i] * chsum[b * DIM + gr * GRP + i];
        gl[t] = s * (1.0f / (float)HW);
    }
    __syncthreads();
    if (t < NB * 24) {
        int b = t / 24, j = t % 24;
        float s = 0.f;
        #pragma unroll
        for (int gr = 0; gr < CR; ++gr) s += gl[b * CR + gr] * cw[j * CR + gr];
        s = (s - bnm[j]) * rsqrtf(bnv[j] + 1e-5f) * bnw[j] + bnb[j];
        fl[t] = 1.0f / (1.0f + expf(-s));
    }
    __syncthreads();
    // A_b[d][k] = gamma[d] * sum_c proj_w[d][c] * F_b[c][k], padded to 112 rows
    for (int idx = t; idx < NB * 112 * CR; idx += 256) {
        int b = idx / (112 * CR);
        int r = idx % (112 * CR);
        int d = r >> 3, k = r & 7;
        float a = 0.f;
        if (d < DIM) {
            const float* f = fl + b * 24;
            float s = 0.f;
            #pragma unroll
            for (int c = 0; c < CR; ++c) {
                float w = 0.f;
                if (k == c) w += f[8 + c];                       // center tap
                int ml = (c == 0) ? 1 : c - 1;                   // reflect left
                if (k == ml) w += f[c];
                int mr = (c == 7) ? 6 : c + 1;                   // reflect right
                if (k == mr) w += f[16 + c];
                s += pw[d * CR + c] * w;
            }
            a = gamma[d] * s;
        }
        Apad[idx] = a;
    }
}

// ------------- Kernel 3: main fused pass with V_WMMA_F32_16X16X4_F32 --------
__global__ __launch_bounds__(256) void main_kernel(const float* __restrict__ x,
                                                   const float* __restrict__ dw,
                                                   const float* __restrict__ beta,
                                                   const float* __restrict__ Apad,
                                                   float* __restrict__ out) {
    __shared__ __align__(16) float smem[DIM * XS + TPIX * CR];
    float* xls = smem;                 // x tile   [104][132]
    float* xrt = smem + DIM * XS;      // xr^T     [128][8]

    const int bx = blockIdx.x;
    const int bb = bx >> 9;            // batch  (512 tiles per image)
    const int tile = bx & 511;
    const size_t P = (size_t)tile * TPIX;
    const float* xb = x + (size_t)bb * DIM * HW;
    const int tid = threadIdx.x;

    // ---- stage 104x128 x-tile into LDS ----
#ifdef HAVE_ASYNC_LDS
    // CDNA5 async copy: global -> LDS directly, tracked by ASYNCcnt.
    for (int i = tid; i < DIM * (TPIX / 4); i += 256) {
        int c = i >> 5;
        int p4 = (i & 31) << 2;
        as1_v4i* gsrc = (as1_v4i*)(uintptr_t)(xb + (size_t)c * HW + P + p4);
        as3_v4i* ldst = (as3_v4i*)(uint32_t)(uintptr_t)(&xls[c * XS + p4]);
        __builtin_amdgcn_global_load_async_to_lds_b128(gsrc, ldst, 0, 0);
    }
    __builtin_amdgcn_s_wait_asynccnt(0);
#else
    for (int i = tid; i < DIM * (TPIX / 4); i += 256) {
        int c = i >> 5;
        int p4 = (i & 31) << 2;
        float4 v = *reinterpret_cast<const float4*>(xb + (size_t)c * HW + P + p4);
        *reinterpret_cast<float4*>(&xls[c * XS + p4]) = v;
    }
#endif
    __syncthreads();

    // ---- grouped 1x1 conv -> xr, stored transposed [pixel][8] ----
    #pragma unroll
    for (int it = 0; it < 4; ++it) {
        int idx = tid + (it << 8);
        int c = idx >> 7, p = idx & (TPIX - 1);
        float s = 0.f;
        #pragma unroll
        for (int i = 0; i < GRP; ++i)
            s += dw[c * GRP + i] * xls[(c * GRP + i) * XS + p];
        xrt[p * CR + c] = s;
    }
    __syncthreads();

    // ---- per-wave WMMA: 16 pixels x 112 output rows ----
    const int lane = tid & 31;
    const int po = (tid >> 5) << 4;           // wave's 16-pixel slice
    const int n = lane & 15;                  // N index (pixel within slice)
    const int k0 = (lane >> 4) << 1;          // K pair: lanes 0-15 -> K0,K1; 16-31 -> K2,K3
    const int mh = (lane >> 4) << 3;          // C/D row offset (+8 for upper half-lanes)
    const int p = po + n;

    // B fragments: B[k][n] = xr[k][pixel]
    v2f blo, bhi;
    blo[0] = xrt[p * CR + k0];     blo[1] = xrt[p * CR + k0 + 1];
    bhi[0] = xrt[p * CR + 4 + k0]; bhi[1] = xrt[p * CR + 5 + k0];

    const float* Ab = Apad + bb * 112 * CR;

    #pragma unroll
    for (int t = 0; t < MT; ++t) {
        const int row = t * 16 + n;           // A-matrix M index = lane&15
        v2f alo, ahi;
        alo[0] = Ab[row * CR + k0];     alo[1] = Ab[row * CR + k0 + 1];
        ahi[0] = Ab[row * CR + 4 + k0]; ahi[1] = Ab[row * CR + 5 + k0];

        // seed accumulator with residual beta[d] * x[d][pixel]
        v8f acc;
        #pragma unroll
        for (int r = 0; r < 8; ++r) {
            int d = t * 16 + r + mh;
            acc[r] = (d < DIM) ? beta[d] * xls[d * XS + p] : 0.f;
        }

        acc = __builtin_amdgcn_wmma_f32_16x16x4_f32(false, alo, false, blo,
                                                    (short)0, acc, false, false);
        acc = __builtin_amdgcn_wmma_f32_16x16x4_f32(false, ahi, false, bhi,
                                                    (short)0, acc, false, false);

        #pragma unroll
        for (int r = 0; r < 8; ++r) {
            int d = t * 16 + r + mh;
            if (d < DIM)
                out[((size_t)bb * DIM + d) * HW + P + p] = acc[r];
        }
    }
}

extern "C" void kernel_launch(void* const* d_in, const int* in_sizes, int n_in,
                              void* d_out, int out_size, void* d_ws, size_t ws_size,
                              hipStream_t stream) {
    const float* x     = (const float*)d_in[0];
    const float* dw_w  = (const float*)d_in[1];
    const float* cw    = (const float*)d_in[2];
    const float* bnw   = (const float*)d_in[3];
    const float* bnb   = (const float*)d_in[4];
    const float* bnm   = (const float*)d_in[5];
    const float* bnv   = (const float*)d_in[6];
    const float* pw    = (const float*)d_in[7];
    const float* gamma = (const float*)d_in[8];
    const float* beta  = (const float*)d_in[9];
    float* out = (float*)d_out;

    float* ws    = (float*)d_ws;
    float* chsum = ws;            // 832 floats
    float* Apad  = ws + 1024;     // 8*112*8 = 7168 floats

    chsum_kernel<<<NB * DIM, 256, 0, stream>>>(x, chsum);
    filt_kernel<<<1, 256, 0, stream>>>(chsum, dw_w, cw, bnw, bnb, bnm, bnv, pw, gamma, Apad);
    main_kernel<<<NB * (HW / TPIX), 256, 0, stream>>>(x, dw_w, beta, Apad, out);
}